// FlashAttentionWithAccumulation_52767968199371
// MI455X (gfx1250) — compile-verified
//
#include <hip/hip_runtime.h>
#include <math.h>

// Problem constants (match reference setup_inputs)
#define B_   2
#define H_   16
#define N_   4096
#define D_   64
#define BK   128     // key block (BLOCK in reference, semantics-critical)
#define BQ   128     // q rows per workgroup
#define NWAVE 8      // 8 waves * 32 lanes = 256 threads
#define KSTR 68      // LDS row stride (floats) for K tile (padded, conflict-free)
#define VSTR 64      // LDS row stride for V tile
#define PSTR 20      // per-wave P-transpose scratch row stride

typedef float v2f __attribute__((ext_vector_type(2)));
typedef float v8f __attribute__((ext_vector_type(8)));
typedef unsigned int v4u __attribute__((ext_vector_type(4)));
typedef int v8i __attribute__((ext_vector_type(8)));
typedef int v4i __attribute__((ext_vector_type(4)));

#if defined(__has_builtin)
#if __has_builtin(__builtin_amdgcn_tensor_load_to_lds) && __has_builtin(__builtin_amdgcn_s_wait_tensorcnt)
#define USE_TDM 1
#endif
#endif

__global__ __launch_bounds__(256)
void fa_accum_kernel(const float* __restrict__ Q, const float* __restrict__ K,
                     const float* __restrict__ V, float* __restrict__ O,
                     float* __restrict__ Cum)
{
  __shared__ float Ks[BK * KSTR];          // 34816 B
  __shared__ float Vs[BK * VSTR];          // 32768 B
  __shared__ float Ps[NWAVE][16 * PSTR];   // 10240 B

  const int qb   = blockIdx.x;
  const int h    = blockIdx.y;
  const int b    = blockIdx.z;
  const int bh   = b * H_ + h;
  const int tid  = threadIdx.x;
  const int wave = tid >> 5;
  const int lane = tid & 31;
  const int half = lane >> 4;     // 0: lanes 0-15, 1: lanes 16-31
  const int lm   = lane & 15;

  const float* qptr = Q + (size_t)bh * N_ * D_;
  const float* kptr = K + (size_t)bh * N_ * D_;
  const float* vptr = V + (size_t)bh * N_ * D_;
  float* optr = O + (size_t)bh * N_ * D_;
  float* cptr = Cum + (size_t)bh * N_;

  const int r0 = qb * BQ + wave * 16;      // first q row owned by this wave

  // ---- Q A-fragments (16x4 tiles along D), pre-scaled by 1/sqrt(D)=0.125 (exact pow2)
  // ISA layout: lanes 0-15 hold M=lane, K={0,1}; lanes 16-31 hold M=lane-16, K={2,3}
  v2f aQ[16];
#pragma unroll
  for (int t = 0; t < 16; ++t) {
    const float* qs = qptr + (size_t)(r0 + lm) * D_ + 4 * t + 2 * half;
    aQ[t].x = qs[0] * 0.125f;
    aQ[t].y = qs[1] * 0.125f;
  }

  // O accumulators: 16x64 = 4 d-tiles of 16x16 (C-layout: row = v + 8*half, col = lm)
  v8f o[4];
  float m_i[8], l_i[8];
#pragma unroll
  for (int dt = 0; dt < 4; ++dt) o[dt] = (v8f)0.0f;
#pragma unroll
  for (int v = 0; v < 8; ++v) { m_i[v] = -INFINITY; l_i[v] = 0.0f; }

  for (int kb = 0; kb < N_ / BK; ++kb) {
    __syncthreads();   // previous iteration's LDS consumers done

#ifdef USE_TDM
    if (wave == 0) {
      // ---- K tile: 128 rows x 64 floats, TDM-padded to 68-float LDS rows ----
      unsigned long long gaK =
          (unsigned long long)(const void*)(kptr + (size_t)kb * BK * D_);
      v4u g0k;
      g0k.x = 1u;                                        // count=1, user descriptor
      g0k.y = (unsigned)(uintptr_t)(void*)Ks;            // lds_addr (bytes)
      g0k.z = (unsigned)(gaK & 0xFFFFFFFFull);           // global_addr[31:0]
      g0k.w = (unsigned)((gaK >> 32) & 0x1FFFFFFull) | (2u << 30); // addr[56:32] | type=2
      v8i g1k;
      // data_size=4B (2<<16) | pad_enable | pad_interval=5 (64 dw) | pad_amount=3 (4 dw)
      g1k[0] = (int)((2u << 16) | (1u << 20) | (5u << 22) | (3u << 25));
      g1k[1] = (int)(64u << 16);            // tensor_dim0 = 64 (bits 79:48 low half)
      g1k[2] = 0;                           // td0 hi | td1 lo16 (131072 -> lo16 = 0)
      g1k[3] = (int)(2u | (64u << 16));     // td1 hi16 = 2 | tile_dim0 = 64
      g1k[4] = BK;                          // tile_dim1 = 128, tile_dim2 = 0
      g1k[5] = 64;                          // tensor_dim0_stride = 64 elements
      g1k[6] = 0;
      g1k[7] = 0;
      v4i zz4 = (v4i)0;
      v8i zz8 = (v8i)0;
      __builtin_amdgcn_tensor_load_to_lds(g0k, g1k, zz4, zz4, zz8, 0);

      // ---- V tile: same tensor, no padding (row-major reads are conflict-benign)
      unsigned long long gaV =
          (unsigned long long)(const void*)(vptr + (size_t)kb * BK * D_);
      v4u g0v;
      g0v.x = 1u;
      g0v.y = (unsigned)(uintptr_t)(void*)Vs;
      g0v.z = (unsigned)(gaV & 0xFFFFFFFFull);
      g0v.w = (unsigned)((gaV >> 32) & 0x1FFFFFFull) | (2u << 30);
      v8i g1v;
      g1v[0] = (int)(2u << 16);
      g1v[1] = (int)(64u << 16);
      g1v[2] = 0;
      g1v[3] = (int)(2u | (64u << 16));
      g1v[4] = BK;
      g1v[5] = 64;
      g1v[6] = 0;
      g1v[7] = 0;
      __builtin_amdgcn_tensor_load_to_lds(g0v, g1v, zz4, zz4, zz8, 0);

      __builtin_amdgcn_s_wait_tensorcnt(0);
    }
#else
    {
      const float* kg = kptr + (size_t)kb * BK * D_;
      const float* vg = vptr + (size_t)kb * BK * D_;
      for (int i = tid; i < BK * (D_ / 4); i += 256) {
        int row = i >> 4;
        int c4  = (i & 15) * 4;
        *(float4*)&Ks[row * KSTR + c4] = *(const float4*)(kg + row * D_ + c4);
        *(float4*)&Vs[row * VSTR + c4] = *(const float4*)(vg + row * D_ + c4);
      }
    }
#endif
    __syncthreads();

    // ---- S = Q * K^T : 8 key sub-tiles of 16, K-dim = D in 16 steps of 4 ----
    v8f S[8];
#pragma unroll
    for (int j = 0; j < 8; ++j) {
      v8f acc = (v8f)0.0f;
#pragma unroll
      for (int t = 0; t < 16; ++t) {
        // B[kk][n] = K[j*16+n][4t+kk]; VGPR0 holds kk=2*half, VGPR1 kk=2*half+1
        const float* kp = &Ks[(j * 16 + lm) * KSTR + 4 * t + 2 * half];
        v2f bK; bK.x = kp[0]; bK.y = kp[1];
        acc = __builtin_amdgcn_wmma_f32_16x16x4_f32(
            false, aQ[t], false, bK, (short)0, acc, false, false);
      }
      S[j] = acc;
    }

    // ---- running row max (reduce over the 16 lanes of this half) ----
    float alpha[8];
#pragma unroll
    for (int v = 0; v < 8; ++v) {
      float mx = S[0][v];
#pragma unroll
      for (int j = 1; j < 8; ++j) mx = fmaxf(mx, S[j][v]);
      mx = fmaxf(mx, __shfl_xor(mx, 8, 32));
      mx = fmaxf(mx, __shfl_xor(mx, 4, 32));
      mx = fmaxf(mx, __shfl_xor(mx, 2, 32));
      mx = fmaxf(mx, __shfl_xor(mx, 1, 32));
      float mnew = fmaxf(m_i[v], mx);
      alpha[v] = expf(m_i[v] - mnew);   // exp(-inf)=0 on first block
      m_i[v] = mnew;
    }

    // ---- P = exp(S - m_running) ----
#pragma unroll
    for (int j = 0; j < 8; ++j) {
#pragma unroll
      for (int v = 0; v < 8; ++v) S[j][v] = expf(S[j][v] - m_i[v]);
    }

    // ---- l update ----
#pragma unroll
    for (int v = 0; v < 8; ++v) {
      float s = 0.0f;
#pragma unroll
      for (int j = 0; j < 8; ++j) s += S[j][v];
      s += __shfl_xor(s, 8, 32);
      s += __shfl_xor(s, 4, 32);
      s += __shfl_xor(s, 2, 32);
      s += __shfl_xor(s, 1, 32);
      l_i[v] = l_i[v] * alpha[v] + s;
    }

    // ---- cumulative received mass per key (atomic across all q blocks) ----
#pragma unroll
    for (int j = 0; j < 8; ++j) {
      float cs = 0.0f;
#pragma unroll
      for (int v = 0; v < 8; ++v) cs += S[j][v];
      cs += __shfl_xor(cs, 16, 32);  // merge rows 0-7 with rows 8-15
      if (half == 0)
        atomicAdd(&cptr[kb * BK + j * 16 + lm], cs);
    }

    // ---- rescale O ----
#pragma unroll
    for (int dt = 0; dt < 4; ++dt) {
#pragma unroll
      for (int v = 0; v < 8; ++v) o[dt][v] *= alpha[v];
    }

    // ---- O += P * V : transpose P tiles via per-wave LDS scratch ----
    float* ps = Ps[wave];
#pragma unroll
    for (int j = 0; j < 8; ++j) {
      // store C-layout: row = v + 8*half, col = lm
#pragma unroll
      for (int v = 0; v < 8; ++v)
        ps[(v + 8 * half) * PSTR + lm] = S[j][v];
      // same-wave LDS ops are in-order; compiler inserts s_wait_dscnt
#pragma unroll
      for (int t = 0; t < 4; ++t) {
        v2f aP;
        aP.x = ps[lm * PSTR + 4 * t + 2 * half];
        aP.y = ps[lm * PSTR + 4 * t + 2 * half + 1];
#pragma unroll
        for (int dt = 0; dt < 4; ++dt) {
          // B[kk][n] = V[j*16 + 4t + kk][dt*16 + n]
          const float* vp = &Vs[(j * 16 + 4 * t + 2 * half) * VSTR + dt * 16 + lm];
          v2f bV; bV.x = vp[0]; bV.y = vp[VSTR];
          o[dt] = __builtin_amdgcn_wmma_f32_16x16x4_f32(
              false, aP, false, bV, (short)0, o[dt], false, false);
        }
      }
    }
  }

  // ---- finalize: out = O / l ----
#pragma unroll
  for (int dt = 0; dt < 4; ++dt) {
#pragma unroll
    for (int v = 0; v < 8; ++v) {
      int row = r0 + v + 8 * half;
      optr[(size_t)row * D_ + dt * 16 + lm] = o[dt][v] / l_i[v];
    }
  }
}

extern "C" void kernel_launch(void* const* d_in, const int* in_sizes, int n_in,
                              void* d_out, int out_size, void* d_ws, size_t ws_size,
                              hipStream_t stream) {
  const float* q = (const float*)d_in[0];
  const float* k = (const float*)d_in[1];
  const float* v = (const float*)d_in[2];
  float* out = (float*)d_out;
  float* cum = out + (size_t)B_ * H_ * N_ * D_;

  // cumulative is accumulated with atomics -> must start at zero every call
  (void)hipMemsetAsync(cum, 0, (size_t)B_ * H_ * N_ * sizeof(float), stream);

  dim3 grid(N_ / BQ, H_, B_);
  fa_accum_kernel<<<grid, 256, 0, stream>>>(q, k, v, out, cum);
}